// Device_Policy_13477607375251
// MI455X (gfx1250) — compile-verified
//
#include <hip/hip_runtime.h>
#include <hip/hip_bf16.h>
#include <math.h>

#define N_NODES      262144
#define NUM_FEAT     64
#define NUM_DEV      64
#define NUM_DEV_FEAT 32
#define H1           128
#define H2           64
#define LEAKY_SLOPE  0.1f
#define EPSF         1e-6f

// kernel-2 tiling
#define KC        64                    // K-chunk per LDS stage
#define KCP       66                    // padded LDS row stride (floats)
#define K2_BLOCKS 128
#define SLAB      (N_NODES / K2_BLOCKS) // 2048
#define CHUNKS    (SLAB / KC)           // 32
// kernel-1 tiling
#define K1_BLOCKS 256
#define K1_ROWS   (N_NODES / K1_BLOCKS) // 1024

typedef float v2f __attribute__((ext_vector_type(2)));
typedef float v8f __attribute__((ext_vector_type(8)));

__device__ __forceinline__ float leaky(float x) { return x >= 0.f ? x : LEAKY_SLOPE * x; }

// ---------------- Kernel 1: column sum / sumsq partials of state ----------------
__global__ void __launch_bounds__(256)
k_state_stats(const float* __restrict__ state,
              float* __restrict__ ssum, float* __restrict__ ssq) {
  const int t   = threadIdx.x;
  const int col = t & 63;
  const int rg  = t >> 6;                 // 0..3
  const int r0  = blockIdx.x * K1_ROWS;
  float s = 0.f, q = 0.f;
  #pragma unroll 4
  for (int i = 0; i < K1_ROWS / 4; ++i) {
    float x = state[(size_t)(r0 + rg + i * 4) * NUM_FEAT + col];
    s += x; q += x * x;
  }
  __shared__ float ls[256], lq[256];
  ls[t] = s; lq[t] = q;
  __syncthreads();
  if (t < 64) {
    ssum[blockIdx.x * 64 + t] = ls[t] + ls[t + 64] + ls[t + 128] + ls[t + 192];
    ssq [blockIdx.x * 64 + t] = lq[t] + lq[t + 64] + lq[t + 128] + lq[t + 192];
  }
}

// ---------------- Kernel 2: mask[64,K] @ mpnn[K,128] partials via WMMA f32 ----
__global__ void __launch_bounds__(256)
k_mask_gemm(const int* __restrict__ assign, const float* __restrict__ mpnn,
            float* __restrict__ partial) {
  __shared__ float maskLds[NUM_DEV * KCP];       // 16.9 KB
  const int t       = threadIdx.x;
  const int lane    = t & 31;
  const int wave    = t >> 5;                    // 0..7  -> N tile
  const int nb      = wave * 16;
  const int halfSel = (lane < 16) ? 0 : 2;       // upper half-wave holds K+2/K+3
  const int l15     = lane & 15;

  v8f acc0 = {0.f,0.f,0.f,0.f,0.f,0.f,0.f,0.f};
  v8f acc1 = acc0, acc2 = acc0, acc3 = acc0;

  const int ldRow  = t >> 2;                     // coop-load: 4 threads per device row
  const int ldQuad = t & 3;
  const int kSlab  = blockIdx.x * SLAB;

  for (int c = 0; c < CHUNKS; ++c) {
    const int k0 = kSlab + c * KC;
    {   // cooperative int->float mask stage into padded LDS
      const int*  src = assign + (size_t)ldRow * N_NODES + k0 + ldQuad * 16;
      float*      dst = &maskLds[ldRow * KCP + ldQuad * 16];
      #pragma unroll
      for (int j = 0; j < 4; ++j) {
        int4 v = ((const int4*)src)[j];
        dst[4 * j + 0] = (v.x == 1) ? 1.f : 0.f;
        dst[4 * j + 1] = (v.y == 1) ? 1.f : 0.f;
        dst[4 * j + 2] = (v.z == 1) ? 1.f : 0.f;
        dst[4 * j + 3] = (v.w == 1) ? 1.f : 0.f;
      }
    }
    __syncthreads();
    // B-fragment base: row (k + halfSel), col nb+l15  (coalesced 64B per half-wave)
    const float* pB = mpnn + (size_t)(k0 + halfSel) * H1 + nb + l15;
    #pragma unroll
    for (int kk = 0; kk < KC; kk += 4) {
      v2f b;
      b.x = pB[(size_t)kk * H1];
      b.y = pB[(size_t)(kk + 1) * H1];
      const int aoff = kk + halfSel;             // even -> 8B aligned ds_load_b64
      v2f a0 = *(const v2f*)&maskLds[(0 * 16 + l15) * KCP + aoff];
      v2f a1 = *(const v2f*)&maskLds[(1 * 16 + l15) * KCP + aoff];
      v2f a2 = *(const v2f*)&maskLds[(2 * 16 + l15) * KCP + aoff];
      v2f a3 = *(const v2f*)&maskLds[(3 * 16 + l15) * KCP + aoff];
      acc0 = __builtin_amdgcn_wmma_f32_16x16x4_f32(false, a0, false, b, (short)0, acc0, false, false);
      acc1 = __builtin_amdgcn_wmma_f32_16x16x4_f32(false, a1, false, b, (short)0, acc1, false, false);
      acc2 = __builtin_amdgcn_wmma_f32_16x16x4_f32(false, a2, false, b, (short)0, acc2, false, false);
      acc3 = __builtin_amdgcn_wmma_f32_16x16x4_f32(false, a3, false, b, (short)0, acc3, false, false);
    }
    __syncthreads();
  }
  // write per-block partial [64][128]; C layout: lane<16 -> M=v, lane>=16 -> M=v+8
  float* pPart   = partial + (size_t)blockIdx.x * (NUM_DEV * H1);
  const int colO = nb + l15;
  const int rAdd = (lane < 16) ? 0 : 8;
  #pragma unroll
  for (int v = 0; v < 8; ++v) {
    pPart[(0 * 16 + v + rAdd) * H1 + colO] = acc0[v];
    pPart[(1 * 16 + v + rAdd) * H1 + colO] = acc1[v];
    pPart[(2 * 16 + v + rAdd) * H1 + colO] = acc2[v];
    pPart[(3 * 16 + v + rAdd) * H1 + colO] = acc3[v];
  }
}

// ---------------- Kernel 3: reductions + normalizations + small embeddings ----
__global__ void __launch_bounds__(256)
k_mid(const float* __restrict__ state, const float* __restrict__ dfs,
      const float* __restrict__ mpnn,
      const float* __restrict__ W1, const float* __restrict__ b1,
      const float* __restrict__ W2, const float* __restrict__ b2,
      const int* __restrict__ predp,
      const float* __restrict__ partial,
      const float* __restrict__ ssum, const float* __restrict__ ssq,
      float* __restrict__ dseN, float* __restrict__ dfe,
      float* __restrict__ lat,  float* __restrict__ rep) {
  __shared__ float meanS[64], invS[64];
  __shared__ float dfm[32], dfi[32];
  __shared__ float dmean[128], dinv[128];
  __shared__ float dse[NUM_DEV * H1];            // 32 KB
  const int t    = threadIdx.x;
  const int pred = *predp;

  if (t < 64) {  // state column stats (ddof=0)
    float S = 0.f, Q = 0.f;
    for (int b = 0; b < K1_BLOCKS; ++b) { S += ssum[b * 64 + t]; Q += ssq[b * 64 + t]; }
    float m   = S / (float)N_NODES;
    float var = Q / (float)N_NODES - m * m;
    var = var > 0.f ? var : 0.f;
    meanS[t] = m;
    invS[t]  = 1.f / (sqrtf(var) + EPSF);
  }
  if (t < 32) {  // device_feat_state column stats
    float S = 0.f, Q = 0.f;
    for (int r = 0; r < NUM_DEV; ++r) { float x = dfs[r * NUM_DEV_FEAT + t]; S += x; Q += x * x; }
    float m   = S / (float)NUM_DEV;
    float var = Q / (float)NUM_DEV - m * m;
    var = var > 0.f ? var : 0.f;
    dfm[t] = m;
    dfi[t] = 1.f / (sqrtf(var) + EPSF);
  }
  // fixed-order reduction of GEMM partials
  for (int o = t; o < NUM_DEV * H1; o += 256) {
    float s = 0.f;
    for (int b = 0; b < K2_BLOCKS; ++b) s += partial[(size_t)b * (NUM_DEV * H1) + o];
    dse[o] = s;
  }
  __syncthreads();
  if (t < 128) {
    // dse column stats over 64 device rows
    float S = 0.f, Q = 0.f;
    for (int r = 0; r < NUM_DEV; ++r) { float x = dse[r * H1 + t]; S += x; Q += x * x; }
    float m   = S / (float)NUM_DEV;
    float var = Q / (float)NUM_DEV - m * m;
    var = var > 0.f ? var : 0.f;
    dmean[t] = m;
    dinv[t]  = 1.f / (sqrtf(var) + EPSF);
    // pred-node latent row: leaky(norm_state[pred] @ W2.T + b2)
    float acc = b2[t];
    const float* w2r = W2 + (size_t)t * NUM_FEAT;
    #pragma unroll 8
    for (int f = 0; f < NUM_FEAT; ++f) {
      float ns = (state[(size_t)pred * NUM_FEAT + f] - meanS[f]) * invS[f];
      acc += ns * w2r[f];
    }
    lat[t] = leaky(acc);
  } else {
    int j = t - 128;
    rep[j] = mpnn[(size_t)pred * H1 + j];
  }
  __syncthreads();
  for (int o = t; o < NUM_DEV * H1; o += 256) {   // normalized device_sum_embedding
    int n = o & 127;
    dseN[o] = (dse[o] - dmean[n]) * dinv[n];
  }
  for (int o = t; o < NUM_DEV * H1; o += 256) {   // device_feat_embedding
    int d = o >> 7, j = o & 127;
    float acc = b1[j];
    const float* w1r = W1 + (size_t)j * NUM_DEV_FEAT;
    #pragma unroll 8
    for (int f = 0; f < NUM_DEV_FEAT; ++f) {
      float x = (dfs[d * NUM_DEV_FEAT + f] - dfm[f]) * dfi[f];
      acc += x * w1r[f];
    }
    dfe[o] = leaky(acc);
  }
}

// ---------------- Kernel 4: head MLP: leaky(concat @ W3.T + b3) @ W4.T + b4 ---
__global__ void __launch_bounds__(64)
k_head(const float* __restrict__ W3, const float* __restrict__ b3,
       const float* __restrict__ W4, const float* __restrict__ b4,
       const float* __restrict__ dfe, const float* __restrict__ lat,
       const float* __restrict__ rep, const float* __restrict__ dseN,
       float* __restrict__ out) {
  const int d = blockIdx.x;
  const int i = threadIdx.x;
  const float* w3r = W3 + (size_t)i * (4 * H1);
  float acc = b3[i];
  #pragma unroll 4
  for (int j = 0; j < H1; ++j) acc += dfe[d * H1 + j] * w3r[j];
  #pragma unroll 4
  for (int j = 0; j < H1; ++j) acc += lat[j] * w3r[H1 + j];
  #pragma unroll 4
  for (int j = 0; j < H1; ++j) acc += rep[j] * w3r[2 * H1 + j];
  #pragma unroll 4
  for (int j = 0; j < H1; ++j) acc += dseN[d * H1 + j] * w3r[3 * H1 + j];
  float h = leaky(acc);
  __shared__ float red[64];
  red[i] = h * W4[i];
  __syncthreads();
  for (int s = 32; s > 0; s >>= 1) {
    if (i < s) red[i] += red[i + s];
    __syncthreads();
  }
  if (i == 0) out[d] = red[0] + b4[0];
}

extern "C" void kernel_launch(void* const* d_in, const int* in_sizes, int n_in,
                              void* d_out, int out_size, void* d_ws, size_t ws_size,
                              hipStream_t stream) {
  const float* state = (const float*)d_in[0];
  const float* dfs   = (const float*)d_in[1];
  const float* mpnn  = (const float*)d_in[2];
  const float* W1    = (const float*)d_in[3];
  const float* b1    = (const float*)d_in[4];
  const float* W2    = (const float*)d_in[5];
  const float* b2    = (const float*)d_in[6];
  const float* W3    = (const float*)d_in[7];
  const float* b3    = (const float*)d_in[8];
  const float* W4    = (const float*)d_in[9];
  const float* b4    = (const float*)d_in[10];
  const int* assign  = (const int*)d_in[11];
  const int* predp   = (const int*)d_in[12];
  float* out = (float*)d_out;

  float* ws      = (float*)d_ws;
  float* partial = ws;                                          // 128*8192 f
  float* ssum    = partial + (size_t)K2_BLOCKS * NUM_DEV * H1;  // 256*64
  float* ssq     = ssum + K1_BLOCKS * 64;                       // 256*64
  float* dseN    = ssq  + K1_BLOCKS * 64;                       // 8192
  float* dfe     = dseN + NUM_DEV * H1;                         // 8192
  float* lat     = dfe  + NUM_DEV * H1;                         // 128
  float* rep     = lat  + H1;                                   // 128

  k_state_stats<<<K1_BLOCKS, 256, 0, stream>>>(state, ssum, ssq);
  k_mask_gemm  <<<K2_BLOCKS, 256, 0, stream>>>(assign, mpnn, partial);
  k_mid        <<<1, 256, 0, stream>>>(state, dfs, mpnn, W1, b1, W2, b2, predp,
                                       partial, ssum, ssq, dseN, dfe, lat, rep);
  k_head       <<<NUM_DEV, 64, 0, stream>>>(W3, b3, W4, b4, dfe, lat, rep, dseN, out);
}